// FAGCN_wodgl_8340826489024
// MI455X (gfx1250) — compile-verified
//
#include <hip/hip_runtime.h>
#include <math.h>

#define N_NODES 50000
#define FIN     512
#define HDIM    128
#define NCLS    40
#define EPS_F   0.3f

typedef __attribute__((ext_vector_type(2))) float v2f;
typedef __attribute__((ext_vector_type(8))) float v8f;

__device__ __forceinline__ void atomAddF(float* p, float v) {
  unsafeAtomicAdd(p, v);
}

// ---------------- elementwise helpers ----------------
__global__ void zero_kernel(float* __restrict__ p, int n) {
  int i = blockIdx.x * blockDim.x + threadIdx.x;
  if (i < n) p[i] = 0.f;
}

__global__ void scale_copy_kernel(const float* __restrict__ src, float* __restrict__ dst,
                                  float s, int n) {
  int i = blockIdx.x * blockDim.x + threadIdx.x;
  if (i < n) dst[i] = s * src[i];
}

// ---------------- degree / dinv ----------------
__global__ void deg_scatter_kernel(const int* __restrict__ col, const float* __restrict__ vals,
                                   float* __restrict__ deg, int E) {
  int e = blockIdx.x * blockDim.x + threadIdx.x;
  if (e < E) atomAddF(&deg[col[e]], vals[e]);
}

__global__ void dinv_kernel(float* __restrict__ deg, int n) {
  int i = blockIdx.x * blockDim.x + threadIdx.x;
  if (i < n) { float d = deg[i]; deg[i] = d > 0.f ? rsqrtf(d) : 0.f; }
}

// ---------------- GEMM1: x = relu(h @ W1 + b1) via fp32 WMMA ----------------
// block = 256 threads = 8 waves; block computes rows [m0, m0+16) x all 128 cols.
// Wave w computes the 16x16 tile at columns [16w, 16w+16). K=512 staged in LDS
// in 128-wide chunks (pitch 132 to avoid bank conflicts).
__global__ __launch_bounds__(256) void gemm1_relu_kernel(const float* __restrict__ h,
                                                         const float* __restrict__ W1,
                                                         const float* __restrict__ b1,
                                                         float* __restrict__ x) {
  __shared__ float As[16][132];
  const int t    = threadIdx.x;
  const int lane = t & 31;
  const int wave = t >> 5;
  const int m0   = blockIdx.x * 16;
  const int n0   = wave * 16;
  const int nn   = n0 + (lane & 15);
  const int kb   = (lane >> 4) << 1;   // 0 for lanes 0-15, 2 for lanes 16-31
  const int mrow = lane & 15;

  v8f c = {};
  for (int kc = 0; kc < FIN; kc += 128) {
    __syncthreads();
    for (int e = t; e < 16 * 128; e += 256) {
      int r = e >> 7, cc = e & 127;
      As[r][cc] = h[(size_t)(m0 + r) * FIN + kc + cc];
    }
    __syncthreads();
#pragma unroll
    for (int kk = 0; kk < 128; kk += 4) {
      // A 16x4 fp32 fragment: VGPR0 = K=kb, VGPR1 = K=kb+1
      v2f a, b;
      a.x = As[mrow][kk + kb];
      a.y = As[mrow][kk + kb + 1];
      // B 4x16 fp32 fragment: row K=kb / K=kb+1 across lanes
      const float* wp = W1 + (size_t)(kc + kk + kb) * HDIM + nn;
      b.x = wp[0];
      b.y = wp[HDIM];
      c = __builtin_amdgcn_wmma_f32_16x16x4_f32(false, a, false, b, (short)0, c,
                                                false, false);
    }
  }
  // C/D layout: VGPR r -> M = r (lanes 0-15) or M = 8+r (lanes 16-31), N = nn
  const int mb = m0 + ((lane >> 4) << 3);
  const float bias = b1[nn];
#pragma unroll
  for (int r = 0; r < 8; ++r) {
    float v = c[r] + bias;
    x[(size_t)(mb + r) * HDIM + nn] = fmaxf(v, 0.f);
  }
}

// ---------------- gate projections: s_src = x@w_src, s_dst = x@w_dst ----------------
// one wave per node, shuffle reduction
__global__ __launch_bounds__(256) void gate_kernel(const float* __restrict__ x,
                                                   const float* __restrict__ gate_w,
                                                   float* __restrict__ s_src,
                                                   float* __restrict__ s_dst, int l) {
  int node = (blockIdx.x * blockDim.x + threadIdx.x) >> 5;
  int lane = threadIdx.x & 31;
  if (node >= N_NODES) return;
  const float* xr   = x + (size_t)node * HDIM;
  const float* wsrc = gate_w + (size_t)l * 2 * HDIM;
  const float* wdst = wsrc + HDIM;
  float a = 0.f, d = 0.f;
#pragma unroll
  for (int j = lane; j < HDIM; j += 32) {
    float xv = xr[j];
    a += xv * wsrc[j];
    d += xv * wdst[j];
  }
#pragma unroll
  for (int off = 16; off; off >>= 1) {
    a += __shfl_xor(a, off, 32);
    d += __shfl_xor(d, off, 32);
  }
  if (lane == 0) { s_src[node] = a; s_dst[node] = d; }
}

// ---------------- edge aggregation ----------------
// one wave per edge: m[col] += 0.5 * tanh(s_src[row]+s_dst[col]+b) * dinv[row]*dinv[col]*vals * x[row]
__global__ __launch_bounds__(256) void edge_agg_kernel(const int* __restrict__ row,
                                                       const int* __restrict__ col,
                                                       const float* __restrict__ vals,
                                                       const float* __restrict__ s_src,
                                                       const float* __restrict__ s_dst,
                                                       const float* __restrict__ dinv,
                                                       const float* __restrict__ gate_b, int l,
                                                       const float* __restrict__ x,
                                                       float* __restrict__ m, int E) {
  int e    = (blockIdx.x * blockDim.x + threadIdx.x) >> 5;
  int lane = threadIdx.x & 31;
  if (e >= E) return;
  int r = row[e], c = col[e];
  float g    = tanhf(s_src[r] + s_dst[c] + gate_b[l]);
  float norm = 0.5f * g * dinv[r] * dinv[c] * vals[e];
  const float4* xr = (const float4*)(x + (size_t)r * HDIM);
  float4 v = xr[lane];                       // 32 lanes x 4 floats = 128
  float* mc = m + (size_t)c * HDIM + lane * 4;
  atomAddF(mc + 0, norm * v.x);
  atomAddF(mc + 1, norm * v.y);
  atomAddF(mc + 2, norm * v.z);
  atomAddF(mc + 3, norm * v.w);
}

// ---------------- GEMM2 + log_softmax ----------------
// one wave per node: lane -> class lane, lanes 0-7 also class lane+32
__global__ __launch_bounds__(256) void gemm2_lsm_kernel(const float* __restrict__ x,
                                                        const float* __restrict__ W2,
                                                        const float* __restrict__ b2,
                                                        float* __restrict__ out) {
  __shared__ float xs[8][HDIM];
  int lane = threadIdx.x & 31;
  int wave = threadIdx.x >> 5;
  int n = blockIdx.x * 8 + wave;
  const float* xr = x + (size_t)n * HDIM;
  for (int j = lane; j < HDIM; j += 32) xs[wave][j] = xr[j];
  __syncthreads();
  int c0 = lane;
  int c1 = lane + 32;
  bool has1 = (c1 < NCLS);
  float v0 = b2[c0];
  float v1 = has1 ? b2[c1] : 0.f;
  for (int k = 0; k < HDIM; ++k) {
    float xv = xs[wave][k];
    v0 += xv * W2[k * NCLS + c0];
    if (has1) v1 += xv * W2[k * NCLS + c1];
  }
  float mx = fmaxf(v0, has1 ? v1 : -INFINITY);
#pragma unroll
  for (int off = 16; off; off >>= 1) mx = fmaxf(mx, __shfl_xor(mx, off, 32));
  float e = expf(v0 - mx) + (has1 ? expf(v1 - mx) : 0.f);
#pragma unroll
  for (int off = 16; off; off >>= 1) e += __shfl_xor(e, off, 32);
  float lse = mx + logf(e);
  out[(size_t)n * NCLS + c0] = v0 - lse;
  if (has1) out[(size_t)n * NCLS + c1] = v1 - lse;
}

extern "C" void kernel_launch(void* const* d_in, const int* in_sizes, int n_in,
                              void* d_out, int out_size, void* d_ws, size_t ws_size,
                              hipStream_t stream) {
  const float* h        = (const float*)d_in[0];
  const int*   row_hom  = (const int*)d_in[1];
  const int*   col_hom  = (const int*)d_in[2];
  const float* vals_hom = (const float*)d_in[3];
  const int*   row_het  = (const int*)d_in[4];
  const int*   col_het  = (const int*)d_in[5];
  const float* vals_het = (const float*)d_in[6];
  const float* W1       = (const float*)d_in[7];
  const float* b1       = (const float*)d_in[8];
  const float* W2       = (const float*)d_in[9];
  const float* b2       = (const float*)d_in[10];
  const float* gate_w   = (const float*)d_in[11];
  const float* gate_b   = (const float*)d_in[12];
  float* out = (float*)d_out;

  const int N = N_NODES;
  const int E = in_sizes[1];
  const int NH = N * HDIM;

  float* ws       = (float*)d_ws;
  float* buf0     = ws;                 // raw = relu(h@W1+b1)
  float* buf1     = buf0 + (size_t)NH;  // layer-1 output
  float* buf2     = buf1 + (size_t)NH;  // layer-2 output
  float* dinv_hom = buf2 + (size_t)NH;  // N floats (deg then dinv, in place)
  float* dinv_het = dinv_hom + N;       // N floats
  float* s_src    = dinv_het + N;       // N floats
  float* s_dst    = s_src + N;          // N floats

  const int T = 256;

  // degree -> dinv (x-independent, once)
  zero_kernel<<<(2 * N + T - 1) / T, T, 0, stream>>>(dinv_hom, 2 * N);
  deg_scatter_kernel<<<(E + T - 1) / T, T, 0, stream>>>(col_hom, vals_hom, dinv_hom, E);
  deg_scatter_kernel<<<(E + T - 1) / T, T, 0, stream>>>(col_het, vals_het, dinv_het, E);
  dinv_kernel<<<(2 * N + T - 1) / T, T, 0, stream>>>(dinv_hom, 2 * N);

  // x = relu(h @ W1 + b1)   (WMMA fp32)
  gemm1_relu_kernel<<<N / 16, T, 0, stream>>>(h, W1, b1, buf0);

  // two FALayers
  const float* xcur = buf0;
  float* bufs[2] = { buf1, buf2 };
  const int edge_blocks = (int)(((size_t)E * 32 + T - 1) / T);
  for (int l = 0; l < 2; ++l) {
    float* mbuf = bufs[l];
    gate_kernel<<<(N * 32 + T - 1) / T, T, 0, stream>>>(xcur, gate_w, s_src, s_dst, l);
    scale_copy_kernel<<<(NH + T - 1) / T, T, 0, stream>>>(buf0, mbuf, EPS_F, NH);
    edge_agg_kernel<<<edge_blocks, T, 0, stream>>>(row_hom, col_hom, vals_hom, s_src, s_dst,
                                                   dinv_hom, gate_b, l, xcur, mbuf, E);
    edge_agg_kernel<<<edge_blocks, T, 0, stream>>>(row_het, col_het, vals_het, s_src, s_dst,
                                                   dinv_het, gate_b, l, xcur, mbuf, E);
    xcur = mbuf;
  }

  // logits + log_softmax
  gemm2_lsm_kernel<<<N / 8, T, 0, stream>>>(xcur, W2, b2, out);
}